// SpectralConv1d_19301583028791
// MI455X (gfx1250) — compile-verified
//
#include <hip/hip_runtime.h>
#include <math.h>

// ---------------------------------------------------------------------------
// SpectralConv1d via Hartley transforms, fp32 end-to-end on V_WMMA_F32_16X16X4_F32
//
// Shapes: x[4,16,8192] f32, w[16,16,64] f32, out[4,16,4097] f32
// Algebra: only the first 64 cols of the big DHT matter, and the final DHT
// input is zero except its first 64 cols -> skinny fp32 GEMMs on the WMMA pipe.
// The K=8192 GEMM uses split-K (32 chunks -> 512 waves) with a deterministic
// two-stage reduction to fill the MI455X instead of 16 serial waves.
// ---------------------------------------------------------------------------

typedef __attribute__((ext_vector_type(2))) float v2f;
typedef __attribute__((ext_vector_type(8))) float v8f;

// ---- workspace layout (in floats) ----
static constexpr int CASN_OFF  = 0;            // cas(8192) cols 0..63 : [8192][64]
static constexpr int CAS64_OFF = 524288;       // cas(64)              : [64][64]
static constexpr int CASF_OFF  = 528384;       // cas(4097) rows 0..63 : [64][4112] (padded stride)
static constexpr int XH_OFF    = 791552;       // xh  [64][64]
static constexpr int HP_OFF    = 795648;       // hp  [64][64]
static constexpr int HQ_OFF    = 799744;       // hq  [256][64]
static constexpr int HZ_OFF    = 816128;       // hz  [64][64]
static constexpr int Z_OFF     = 820224;       // z   [64][64]
static constexpr int PART_OFF  = 824320;       // split-K partials [32][64][64]

static constexpr int SPLITK_CHUNKS = 32;       // 8192 / 32 = 256 K per chunk

// cas(N)[r,c] = cos(a) + sin(a), a = (2*pi/N) * (r*c), with the product rounded
// in fp32 exactly like jnp.outer(n,n) in float32 (mimics reference numerics).
__global__ void gen_cas_kernel(float* __restrict__ out, int rows, int cols,
                               int stride, float scale) {
  int idx = blockIdx.x * blockDim.x + threadIdx.x;
  if (idx >= rows * cols) return;
  int r = idx / cols;
  int c = idx - r * cols;
  float prod = (float)r * (float)c;   // fp32 rounding, as in the fp32 outer()
  float ang  = scale * prod;
  float s, cs;
  sincosf(ang, &s, &cs);
  out[r * stride + c] = cs + s;
}

// Generic fp32 WMMA GEMM: C[M x Ncols] = scale * (A[M x K] @ B[K x N-ish])
// One wave (32 threads) per 16x16 output tile; K stepped by 4 via
// v_wmma_f32_16x16x4_f32.  M multiple of 16, K multiple of 4; B may be read
// past Ncols (padded stride), stores guarded by col < Ncols.
//
// VGPR layouts (ISA 7.12.2, wave32):
//   A 16x4 f32 : lanes 0-15 -> {K0,K1} of row M=lane; lanes 16-31 -> {K2,K3}
//   B 4x16 f32 : mirrored (lane half selects K pair, lane%16 selects N)
//   C/D 16x16  : VGPR r -> M = r + 8*(lane>=16), N = lane%16
__global__ void wmma_gemm_f32_kernel(const float* __restrict__ A, int lda,
                                     const float* __restrict__ B, int ldb,
                                     float* __restrict__ C, int ldc,
                                     int K, int Ncols, float scale, int ntilesN) {
  const int t    = blockIdx.x;
  const int tm   = t / ntilesN;
  const int tn   = t - tm * ntilesN;
  const int lane = threadIdx.x & 31;
  const int half = lane >> 4;       // 0: K pair {0,1}, 1: K pair {2,3}
  const int lm   = lane & 15;

  const int arow = tm * 16 + lm;
  const int bcol = tn * 16 + lm;

  v8f acc = {};
  for (int k0 = 0; k0 < K; k0 += 4) {
    const int ka = k0 + 2 * half;
    v2f a;
    a.x = A[arow * lda + ka];
    a.y = A[arow * lda + ka + 1];
    v2f b;
    b.x = B[ka * ldb + bcol];
    b.y = B[(ka + 1) * ldb + bcol];
    acc = __builtin_amdgcn_wmma_f32_16x16x4_f32(
        false, a, false, b, (short)0, acc, false, false);
  }

  const int col = tn * 16 + lm;
#pragma unroll
  for (int r = 0; r < 8; ++r) {
    const int row = tm * 16 + half * 8 + r;
    if (col < Ncols) C[row * ldc + col] = acc[r] * scale;
  }
}

// Split-K WMMA GEMM for the K=8192 stage: grid = ntiles * SPLITK_CHUNKS waves.
// Each wave computes a 16x16 partial over its K chunk and writes it to
// part[chunk][64][64].  M=N=64 fixed (4x4 tiles).
__global__ void wmma_gemm_splitk_kernel(const float* __restrict__ A, int lda,
                                        const float* __restrict__ B, int ldb,
                                        float* __restrict__ part, int kChunk) {
  const int t     = blockIdx.x;
  const int tile  = t / SPLITK_CHUNKS;
  const int chunk = t - tile * SPLITK_CHUNKS;
  const int tm    = tile >> 2;
  const int tn    = tile & 3;
  const int lane  = threadIdx.x & 31;
  const int half  = lane >> 4;
  const int lm    = lane & 15;

  const int arow  = tm * 16 + lm;
  const int bcol  = tn * 16 + lm;
  const int kbeg  = chunk * kChunk;
  const int kend  = kbeg + kChunk;

  v8f acc = {};
  for (int k0 = kbeg; k0 < kend; k0 += 4) {
    const int ka = k0 + 2 * half;
    v2f a;
    a.x = A[arow * lda + ka];
    a.y = A[arow * lda + ka + 1];
    v2f b;
    b.x = B[ka * ldb + bcol];
    b.y = B[(ka + 1) * ldb + bcol];
    acc = __builtin_amdgcn_wmma_f32_16x16x4_f32(
        false, a, false, b, (short)0, acc, false, false);
  }

  float* dst = part + chunk * 4096;   // [64][64] slice for this chunk
  const int col = tn * 16 + lm;
#pragma unroll
  for (int r = 0; r < 8; ++r) {
    const int row = tm * 16 + half * 8 + r;
    dst[row * 64 + col] = acc[r];
  }
}

// Deterministic reduction of split-K partials: xh[idx] = sum_c part[c][idx]
// (fixed summation order -> bitwise-identical across graph replays).
__global__ void reduce_partials_kernel(const float* __restrict__ part,
                                       float* __restrict__ xh) {
  int idx = blockIdx.x * blockDim.x + threadIdx.x;   // 4096
  if (idx >= 4096) return;
  float s = 0.0f;
#pragma unroll 8
  for (int c = 0; c < SPLITK_CHUNKS; ++c) s += part[c * 4096 + idx];
  xh[idx] = s;
}

// hz[b,o,m] = 0.5 * sum_i ( (hp[b,i,m]+hp[b,i,mr]) * hq[i,o,m]
//                         + (hp[b,i,m]-hp[b,i,mr]) * hq[i,o,mr] ),  mr=(64-m)%64
__global__ void hz_kernel(const float* __restrict__ hp,
                          const float* __restrict__ hq,
                          float* __restrict__ hz) {
  int idx = blockIdx.x * blockDim.x + threadIdx.x;   // 4096 = 4*16*64
  if (idx >= 4096) return;
  const int m  = idx & 63;
  const int o  = (idx >> 6) & 15;
  const int b  = idx >> 10;
  const int mr = (64 - m) & 63;
  float s = 0.0f;
#pragma unroll 4
  for (int i = 0; i < 16; ++i) {
    const float p  = hp[(b * 16 + i) * 64 + m];
    const float pr = hp[(b * 16 + i) * 64 + mr];
    const float q  = hq[(i * 16 + o) * 64 + m];
    const float qr = hq[(i * 16 + o) * 64 + mr];
    s += (p + pr) * q + (p - pr) * qr;
  }
  hz[(b * 16 + o) * 64 + m] = 0.5f * s;
}

extern "C" void kernel_launch(void* const* d_in, const int* in_sizes, int n_in,
                              void* d_out, int out_size, void* d_ws, size_t ws_size,
                              hipStream_t stream) {
  (void)in_sizes; (void)n_in; (void)out_size; (void)ws_size;

  const float* x = (const float*)d_in[0];   // [4,16,8192] -> [64 x 8192]
  const float* w = (const float*)d_in[1];   // [16,16,64]  -> [256 x 64]
  float* out = (float*)d_out;               // [4,16,4097] -> [64 x 4097]
  float* ws  = (float*)d_ws;

  float* casN  = ws + CASN_OFF;
  float* cas64 = ws + CAS64_OFF;
  float* casF  = ws + CASF_OFF;
  float* xh    = ws + XH_OFF;
  float* hp    = ws + HP_OFF;
  float* hq    = ws + HQ_OFF;
  float* hz    = ws + HZ_OFF;
  float* z     = ws + Z_OFF;
  float* part  = ws + PART_OFF;

  const double TWO_PI = 6.283185307179586476925286766559;

  // 1) cas tables
  gen_cas_kernel<<<(8192 * 64 + 255) / 256, 256, 0, stream>>>(
      casN, 8192, 64, 64, (float)(TWO_PI / 8192.0));
  gen_cas_kernel<<<(64 * 64 + 255) / 256, 256, 0, stream>>>(
      cas64, 64, 64, 64, (float)(TWO_PI / 64.0));
  gen_cas_kernel<<<(64 * 4112 + 255) / 256, 256, 0, stream>>>(
      casF, 64, 4112, 4112, (float)(TWO_PI / 4097.0));

  // 2) xh[64x64] = x[64x8192] @ casN[8192x64], split-K (16 tiles x 32 chunks)
  wmma_gemm_splitk_kernel<<<16 * SPLITK_CHUNKS, 32, 0, stream>>>(
      x, 8192, casN, 64, part, 8192 / SPLITK_CHUNKS);
  reduce_partials_kernel<<<16, 256, 0, stream>>>(part, xh);

  // 3) hp[64x64] = xh @ cas64
  wmma_gemm_f32_kernel<<<4 * 4, 32, 0, stream>>>(
      xh, 64, cas64, 64, hp, 64, 64, 64, 1.0f, 4);

  // 4) hq[256x64] = w @ cas64
  wmma_gemm_f32_kernel<<<16 * 4, 32, 0, stream>>>(
      w, 64, cas64, 64, hq, 64, 64, 64, 1.0f, 4);

  // 5) hz (Hartley cross-term over Cin)
  hz_kernel<<<16, 256, 0, stream>>>(hp, hq, hz);

  // 6) z[64x64] = hz @ cas64 / (B*Cout*M = 4096)
  wmma_gemm_f32_kernel<<<4 * 4, 32, 0, stream>>>(
      hz, 64, cas64, 64, z, 64, 64, 64, 1.0f / 4096.0f, 4);

  // 7) out[64x4097] = z @ casF / (B*Cout*N2 = 262208); N padded to 4112 tiles
  wmma_gemm_f32_kernel<<<4 * 257, 32, 0, stream>>>(
      z, 64, casF, 4112, out, 4097, 64, 4097, 1.0f / 262208.0f, 257);
}